// TelephotoInterp_77764677861766
// MI455X (gfx1250) — compile-verified
//
#include <hip/hip_runtime.h>
#include <stdint.h>
#include <cmath>

#define TILE  2048
#define BLOCK 256

typedef unsigned int u32;
typedef u32   u32x4 __attribute__((ext_vector_type(4)));
typedef int   i32x4 __attribute__((ext_vector_type(4)));
typedef int   i32x8 __attribute__((ext_vector_type(8)));
typedef float f32x2 __attribute__((ext_vector_type(2)));
typedef float f32x8 __attribute__((ext_vector_type(8)));

#if __has_include(<hip/amd_detail/amd_gfx1250_TDM.h>)
#define TDM_6ARG 1
#else
#define TDM_6ARG 0
#endif

#if defined(__AMDGCN__) && __has_builtin(__builtin_amdgcn_wmma_f32_16x16x4_f32)
#define HAVE_WMMA 1
#else
#define HAVE_WMMA 0
#endif

// -------- Tensor Data Mover: 1-D dword copy global -> LDS ------------------
// D# descriptor per CDNA5 ISA §8.3/8.4: count=1, type=2, data_size=4B,
// 1-D tile of `ndw` dwords. Words forced into SGPRs via readfirstlane.
__device__ __forceinline__ void tdm_load_1d(const void* gaddr, u32 lds_byte, u32 ndw) {
    u32 alo = (u32)__builtin_amdgcn_readfirstlane((int)(u32)(uintptr_t)gaddr);
    u32 ahi = (u32)__builtin_amdgcn_readfirstlane((int)(u32)((uintptr_t)gaddr >> 32));
    u32 lds = (u32)__builtin_amdgcn_readfirstlane((int)lds_byte);
    u32 nd  = (u32)__builtin_amdgcn_readfirstlane((int)ndw);

    u32x4 g0;
    g0.x = 1u;                                    // count=1, user mode, no gather
    g0.y = lds;                                   // lds_addr (bytes)
    g0.z = alo;                                   // global_addr[31:0]
    g0.w = (ahi & 0x01FFFFFFu) | (2u << 30);      // global_addr[56:32] | type=2

    i32x8 g1;
    g1[0] = (int)(2u << 16);                      // wg_mask=0 | data_size=2 (4B)
    g1[1] = (int)((nd & 0xFFFFu) << 16);          // tensor_dim0[15:0]
    g1[2] = (int)(((nd >> 16) & 0xFFFFu) | (1u << 16)); // tensor_dim0[31:16] | tensor_dim1=1
    g1[3] = (int)((nd & 0xFFFFu) << 16);          // tile_dim0 = nd
    g1[4] = 0;                                    // tile_dim1=0, tile_dim2=0 (1-D)
    g1[5] = (int)nd;                              // tensor_dim0_stride[31:0]
    g1[6] = (int)((nd >> 16) & 0xFFFFu);          // stride0[47:32] | stride1[15:0]=0
    g1[7] = 0;

    i32x4 gz = {0, 0, 0, 0};
#if TDM_6ARG
    i32x8 gz8 = {0, 0, 0, 0, 0, 0, 0, 0};
    __builtin_amdgcn_tensor_load_to_lds(g0, g1, gz, gz, gz8, 0);
#else
    __builtin_amdgcn_tensor_load_to_lds(g0, g1, gz, gz, 0);
#endif
}

// -------- grid zeroing -----------------------------------------------------
__global__ void __launch_bounds__(256) zero_kernel(float* __restrict__ out, int n) {
    int i = 4 * (blockIdx.x * blockDim.x + threadIdx.x);
    if (i + 3 < n) {
        float4 z = make_float4(0.f, 0.f, 0.f, 0.f);
        *(float4*)(out + i) = z;
    } else {
        for (int j = i; j < n; ++j) out[j] = 0.f;
    }
}

// -------- main paint kernel ------------------------------------------------
__global__ void __launch_bounds__(BLOCK) paint_kernel(
    const float* __restrict__ pos, const float* __restrict__ vel,
    const float* __restrict__ rot, const float* __restrict__ obs,
    const float* __restrict__ rcen, const float* __restrict__ dwid,
    const float* __restrict__ t_p, const float* __restrict__ maxd_p,
    const float* __restrict__ box_p, const int* __restrict__ shell_p,
    const int* __restrict__ rotidx_p, float* __restrict__ out,
    int N, int numTiles, int res)
{
    __shared__ float s_pos[2][TILE * 3];
    __shared__ float s_vel[2][TILE * 3];

    // Uniform scalar parameters.
    const int   shell    = shell_p[0];
    const float r_center = rcen[shell];
    const float width    = dwid[shell];
    const float maxd     = maxd_p[0];
    const float box      = box_p[0];
    const float t        = t_p[0];
    const bool  inside   = (r_center + 0.5f * width) <= maxd;
    const float ox = obs[0], oy = obs[1], oz = obs[2];
    const int   ridx = ((rotidx_p[0] % 47) + 47) % 47;
    const float* M = rot + ridx * 9;
    const float m00 = M[0], m01 = M[1], m02 = M[2];
    const float m10 = M[3], m11 = M[4], m12 = M[5];
    const float m20 = M[6], m21 = M[7], m22 = M[8];
    const float shift = floorf(r_center / maxd) * maxd;
    const float scale = (float)res / box;
    const float zlo = r_center - 0.5f * width, zhi = r_center + 0.5f * width;
    const float Gp_t = t * sqrtf(t);                  // t^1.5
    const bool  pow2 = (res & (res - 1)) == 0;
    const int   mask = res - 1;

    // CIC paint of one final position (4 no-return global_atomic_add_f32).
    auto paint = [&](float qx, float qy, float qz) {
        if (qz >= zlo && qz < zhi) {
            const float x = qx * scale, y = qy * scale;
            const float xf = floorf(x), yf = floorf(y);
            const float fx = x - xf, fy = y - yf;
            int ix0, iy0;
            if (pow2) { ix0 = ((int)xf) & mask; iy0 = ((int)yf) & mask; }
            else {
                ix0 = (int)xf % res; if (ix0 < 0) ix0 += res;
                iy0 = (int)yf % res; if (iy0 < 0) iy0 += res;
            }
            int ix1 = ix0 + 1; if (ix1 == res) ix1 = 0;
            int iy1 = iy0 + 1; if (iy1 == res) iy1 = 0;
            const float w00 = (1.f - fx) * (1.f - fy);
            const float w10 = fx * (1.f - fy);
            const float w01 = (1.f - fx) * fy;
            const float w11 = fx * fy;
            __hip_atomic_fetch_add(&out[ix0 * res + iy0], w00, __ATOMIC_RELAXED, __HIP_MEMORY_SCOPE_AGENT);
            __hip_atomic_fetch_add(&out[ix1 * res + iy0], w10, __ATOMIC_RELAXED, __HIP_MEMORY_SCOPE_AGENT);
            __hip_atomic_fetch_add(&out[ix0 * res + iy1], w01, __ATOMIC_RELAXED, __HIP_MEMORY_SCOPE_AGENT);
            __hip_atomic_fetch_add(&out[ix1 * res + iy1], w11, __ATOMIC_RELAXED, __HIP_MEMORY_SCOPE_AGENT);
        }
    };

    // Drift correction + paint from already-rotated (r) and rotated-velocity (w).
    auto drift_paint = [&](float rx, float ry, float rz, float wx, float wy, float wz) {
        const float ex = rx - ox, ey = ry - oy, ez = rz - oz;
        const float dist = sqrtf(ex * ex + ey * ey + ez * ez);
        const float a  = 1.0f / (1.0f + dist * (1.0f / 3000.0f));
        const float ac = sqrtf(t * a);
        const float drift = (a * sqrtf(a) - Gp_t) * (1.0f / 1.5f) * rsqrtf(ac);
        paint(rx + drift * wx + ox, ry + drift * wy + oy, rz + drift * wz + oz);
    };

#if HAVE_WMMA
    // Per-lane A-matrix fragments for V_WMMA_F32_16X16X4_F32 (A 16x4 layout:
    // lane m<16 -> (A[m,0],A[m,1]); lane m+16 -> (A[m,2],A[m,3])).
    // A_lo rows 0..2 / A_hi rows 8..10 = [M_r | t_r], t_r = shiftvec_r - (M.obs)_r.
    f32x2 Alo = {0.f, 0.f}, Ahi = {0.f, 0.f};
    {
        const int lane = (int)(threadIdx.x & 31);
        const int n = lane & 15;
        const bool hi = lane >= 16;
        const float t0 = -(m00 * ox + m01 * oy + m02 * oz);
        const float t1 = -(m10 * ox + m11 * oy + m12 * oz);
        const float t2 = shift - (m20 * ox + m21 * oy + m22 * oz);
        if      (n == 0)  { Alo[0] = hi ? m02 : m00; Alo[1] = hi ? t0 : m01; }
        else if (n == 1)  { Alo[0] = hi ? m12 : m10; Alo[1] = hi ? t1 : m11; }
        else if (n == 2)  { Alo[0] = hi ? m22 : m20; Alo[1] = hi ? t2 : m21; }
        if      (n == 8)  { Ahi[0] = hi ? m02 : m00; Ahi[1] = hi ? t0 : m01; }
        else if (n == 9)  { Ahi[0] = hi ? m12 : m10; Ahi[1] = hi ? t1 : m11; }
        else if (n == 10) { Ahi[0] = hi ? m22 : m20; Ahi[1] = hi ? t2 : m21; }
    }
#endif

    // Uniform (SGPR) wave id: TDM ignores EXEC, so the issue gate must be a
    // scalar branch.
    const int wave = __builtin_amdgcn_readfirstlane((int)(threadIdx.x >> 5));

    int tile = blockIdx.x;
    if (tile >= numTiles) return;

    // Prime the TDM pipeline: tile0 -> buffer 0.
    if (wave == 0) {
        int s0 = tile * TILE;
        int c0 = min(TILE, N - s0);
        tdm_load_1d(pos + 3 * (long)s0, (u32)(uintptr_t)&s_pos[0][0], (u32)(3 * c0));
        tdm_load_1d(vel + 3 * (long)s0, (u32)(uintptr_t)&s_vel[0][0], (u32)(3 * c0));
    }

    int it = 0;
    for (; tile < numTiles; tile += (int)gridDim.x, ++it) {
        const int buf = it & 1;
        const int nextTile = tile + (int)gridDim.x;

        if (wave == 0) {
            if (nextTile < numTiles) {
                int ns = nextTile * TILE;
                int nc = min(TILE, N - ns);
                tdm_load_1d(pos + 3 * (long)ns, (u32)(uintptr_t)&s_pos[buf ^ 1][0], (u32)(3 * nc));
                tdm_load_1d(vel + 3 * (long)ns, (u32)(uintptr_t)&s_vel[buf ^ 1][0], (u32)(3 * nc));
                __builtin_amdgcn_s_wait_tensorcnt(2);   // current tile's 2 DMAs done
            } else {
                __builtin_amdgcn_s_wait_tensorcnt(0);   // drain
            }
        }
        __syncthreads();   // publish LDS tile to all 8 waves

        const int start = tile * TILE;
        const int cnt = min(TILE, N - start);

#if HAVE_WMMA
        if (!inside && cnt == TILE) {
            // WMMA path: full tile, uniform branch => EXEC all ones at WMMA.
            // Each wave: 32 particles/step via 4 chained 16x16x4 WMMAs.
            const int lane = (int)(threadIdx.x & 31);
            const int wv   = (int)(threadIdx.x >> 5);
            const int n    = lane & 15;
            const bool hi  = lane >= 16;
            const f32x8 z8 = {0.f, 0.f, 0.f, 0.f, 0.f, 0.f, 0.f, 0.f};
            for (int s = 0; s < TILE / BLOCK; ++s) {
                const int base = s * BLOCK + wv * 32;
                const int iA = base + n;        // particles base .. base+15
                const int iB = base + 16 + n;   // particles base+16 .. base+31
                // B 4x16 layout: lane n<16 -> (B[0,n],B[1,n]) = (x,y);
                //                lane n+16 -> (B[2,n],B[3,n]) = (z,w).
                f32x2 Bp1, Bp2, Bv1, Bv2;
                if (!hi) {
                    Bp1[0] = s_pos[buf][3 * iA + 0]; Bp1[1] = s_pos[buf][3 * iA + 1];
                    Bp2[0] = s_pos[buf][3 * iB + 0]; Bp2[1] = s_pos[buf][3 * iB + 1];
                    Bv1[0] = s_vel[buf][3 * iA + 0]; Bv1[1] = s_vel[buf][3 * iA + 1];
                    Bv2[0] = s_vel[buf][3 * iB + 0]; Bv2[1] = s_vel[buf][3 * iB + 1];
                } else {
                    Bp1[0] = s_pos[buf][3 * iA + 2]; Bp1[1] = 1.0f;  // w=1: affine
                    Bp2[0] = s_pos[buf][3 * iB + 2]; Bp2[1] = 1.0f;
                    Bv1[0] = s_vel[buf][3 * iA + 2]; Bv1[1] = 0.0f;  // w=0: linear
                    Bv2[0] = s_vel[buf][3 * iB + 2]; Bv2[1] = 0.0f;
                }
                // D vgpr r, lane l = coord r of particle base+l (rows 0-2 serve
                // lanes 0-15 via A_lo, rows 8-10 serve lanes 16-31 via A_hi).
                f32x8 dp = __builtin_amdgcn_wmma_f32_16x16x4_f32(false, Alo, false, Bp1, (short)0, z8, false, false);
                dp       = __builtin_amdgcn_wmma_f32_16x16x4_f32(false, Ahi, false, Bp2, (short)0, dp, false, false);
                f32x8 dv = __builtin_amdgcn_wmma_f32_16x16x4_f32(false, Alo, false, Bv1, (short)0, z8, false, false);
                dv       = __builtin_amdgcn_wmma_f32_16x16x4_f32(false, Ahi, false, Bv2, (short)0, dv, false, false);
                drift_paint(dp[0], dp[1], dp[2], dv[0], dv[1], dv[2]);
            }
        } else
#endif
        {
            // Scalar path: inside-box branch, partial tiles, or no-WMMA build.
            for (int p = (int)threadIdx.x; p < cnt; p += BLOCK) {
                const float px = s_pos[buf][3 * p + 0];
                const float py = s_pos[buf][3 * p + 1];
                const float pz = s_pos[buf][3 * p + 2];
                if (inside) {
                    paint(px, py, pz);
                } else {
                    const float vx = s_vel[buf][3 * p + 0];
                    const float vy = s_vel[buf][3 * p + 1];
                    const float vz = s_vel[buf][3 * p + 2];
                    const float dx = px - ox, dy = py - oy, dz = pz - oz;
                    const float rx = m00 * dx + m01 * dy + m02 * dz;
                    const float ry = m10 * dx + m11 * dy + m12 * dz;
                    const float rz = m20 * dx + m21 * dy + m22 * dz + shift;
                    const float wx = m00 * vx + m01 * vy + m02 * vz;
                    const float wy = m10 * vx + m11 * vy + m12 * vz;
                    const float wz = m20 * vx + m21 * vy + m22 * vz;
                    drift_paint(rx, ry, rz, wx, wy, wz);
                }
            }
        }
        __syncthreads();   // tile fully consumed; buffer may be overwritten
    }
}

extern "C" void kernel_launch(void* const* d_in, const int* in_sizes, int n_in,
                              void* d_out, int out_size, void* d_ws, size_t ws_size,
                              hipStream_t stream) {
    const float* pos    = (const float*)d_in[0];
    const float* vel    = (const float*)d_in[1];
    const float* rotm   = (const float*)d_in[2];
    const float* obs    = (const float*)d_in[3];
    const float* rcen   = (const float*)d_in[4];
    const float* dwid   = (const float*)d_in[5];
    const float* t_p    = (const float*)d_in[6];
    const float* maxd_p = (const float*)d_in[7];
    const float* box_p  = (const float*)d_in[8];
    const int*   shell  = (const int*)d_in[9];
    const int*   rotidx = (const int*)d_in[10];
    float*       out    = (float*)d_out;

    const int N   = in_sizes[0] / 3;
    const int res = (int)lround(sqrt((double)out_size));   // 2048 for given setup
    const int numTiles = (N + TILE - 1) / TILE;

    // Zero the output grid (harness poisons d_out).
    const int zthreads = (out_size + 3) / 4;
    zero_kernel<<<(zthreads + 255) / 256, 256, 0, stream>>>(out, out_size);

    // Persistent blocks: each runs ~numTiles/blocks tiles, TDM double-buffered.
    int blocks = numTiles < 1024 ? numTiles : 1024;
    paint_kernel<<<blocks, BLOCK, 0, stream>>>(
        pos, vel, rotm, obs, rcen, dwid, t_p, maxd_p, box_p, shell, rotidx,
        out, N, numTiles, res);
}